// MixtureOfExperts_88742614270301
// MI455X (gfx1250) — compile-verified
//
#include <hip/hip_runtime.h>
#include <math.h>

#define N_TOK 16384
#define DIM   1024
#define NEXP  8

#define MT  32    // tokens per block tile
#define FT  128   // features per block tile (2 x 16 per wave -> A-fragment reuse)
#define KC  32    // K chunk (double-buffered)
#define XST 36    // padded LDS row stride (dwords): 36*4 = 144B = 9*16B -> 16B aligned rows, conflict-free b64 reads

typedef float v2f __attribute__((ext_vector_type(2)));
typedef float v8f __attribute__((ext_vector_type(8)));

// ---- CDNA5 async global->LDS copy (ASYNCcnt path), 16B per lane ----
__device__ __forceinline__ void async_g2l_b128(unsigned lds_off, const float* g) {
    asm volatile("global_load_async_to_lds_b128 %0, %1, off"
                 :: "v"(lds_off), "v"((unsigned long long)(uintptr_t)g)
                 : "memory");
}
__device__ __forceinline__ void wait_async0() {
    asm volatile("s_wait_asynccnt 0x0" ::: "memory");
}

// ---------------- Router: logits -> softmax -> top2 -> scatter to per-expert lists ----------------
__global__ __launch_bounds__(256) void moe_router(
    const float* __restrict__ x, const float* __restrict__ ctx,
    const float* __restrict__ rw, const float* __restrict__ rb,
    const float* __restrict__ cwp,
    int* __restrict__ cnt, float* __restrict__ wsum,
    int* __restrict__ list, float* __restrict__ gw)
{
    __shared__ __attribute__((aligned(16))) float rwS[NEXP * DIM];   // 32 KB
    const int tid = threadIdx.x;

    const float4* rw4  = (const float4*)rw;
    float4*       rwS4 = (float4*)rwS;
    #pragma unroll
    for (int i = 0; i < (NEXP * DIM / 4) / 256; ++i)
        rwS4[tid + 256 * i] = rw4[tid + 256 * i];
    __syncthreads();

    const float cw = cwp[0];
    const int wave = tid >> 5, lane = tid & 31;
    const int base = blockIdx.x * 64 + wave * 8;

    for (int j = 0; j < 8; ++j) {
        const int n = base + j;
        float acc[NEXP];
        #pragma unroll
        for (int e = 0; e < NEXP; ++e) acc[e] = 0.f;

        const float4* xr = (const float4*)(x + (size_t)n * DIM);
        #pragma unroll
        for (int k = 0; k < 8; ++k) {
            const float4 xv = xr[k * 32 + lane];
            const int d = (k * 32 + lane) * 4;
            #pragma unroll
            for (int e = 0; e < NEXP; ++e) {
                acc[e] += xv.x * rwS[e * DIM + d]     + xv.y * rwS[e * DIM + d + 1]
                        + xv.z * rwS[e * DIM + d + 2] + xv.w * rwS[e * DIM + d + 3];
            }
        }
        #pragma unroll
        for (int e = 0; e < NEXP; ++e) {
            float v = acc[e];
            #pragma unroll
            for (int off = 16; off > 0; off >>= 1) v += __shfl_xor(v, off, 32);
            acc[e] = v;
        }

        float logit[NEXP], m = -1e30f;
        #pragma unroll
        for (int e = 0; e < NEXP; ++e) {
            logit[e] = acc[e] + rb[e] + cw * ctx[e];
            m = fmaxf(m, logit[e]);
        }
        float p[NEXP], s = 0.f;
        #pragma unroll
        for (int e = 0; e < NEXP; ++e) { p[e] = expf(logit[e] - m); s += p[e]; }
        const float inv = 1.f / s;
        #pragma unroll
        for (int e = 0; e < NEXP; ++e) p[e] *= inv;

        if (lane < NEXP) atomicAdd(&wsum[lane], p[lane]);

        if (lane == 0) {
            int i0 = 0;
            #pragma unroll
            for (int e = 1; e < NEXP; ++e) if (p[e] > p[i0]) i0 = e;   // first max on ties
            int i1 = (i0 == 0) ? 1 : 0;
            #pragma unroll
            for (int e = 0; e < NEXP; ++e) if (e != i0 && p[e] > p[i1]) i1 = e;
            const float r = 1.f / (p[i0] + p[i1]);
            int pos0 = atomicAdd(&cnt[i0], 1);
            list[i0 * N_TOK + pos0] = n;  gw[i0 * N_TOK + pos0] = p[i0] * r;
            int pos1 = atomicAdd(&cnt[i1], 1);
            list[i1 * N_TOK + pos1] = n;  gw[i1 * N_TOK + pos1] = p[i1] * r;
        }
    }
}

// ---------------- Grouped expert GEMM: double-buffered async LDS pipeline + f32 WMMA ----------------
__global__ __launch_bounds__(256) void moe_expert_gemm(
    const float* __restrict__ x, const float* __restrict__ ew,
    const float* __restrict__ eb, const int* __restrict__ cnt,
    const int* __restrict__ list, const float* __restrict__ gw,
    float* __restrict__ out)
{
    const int e = blockIdx.z;
    const int count = cnt[e];
    const int m0 = blockIdx.y * MT;
    if (m0 >= count) return;                 // data-dependent early exit
    const int f0 = blockIdx.x * FT;

    __shared__ __attribute__((aligned(16))) float Xs[2][MT * XST];   // 2 x 4.5 KB
    __shared__ __attribute__((aligned(16))) float Ws[2][FT * XST];   // 2 x 18  KB
    __shared__ int   tokS[MT];
    __shared__ float wS[MT];

    const int tid = threadIdx.x;
    if (tid < MT) {
        const int idx = m0 + tid;
        const bool v = idx < count;
        tokS[tid] = v ? list[e * N_TOK + idx] : 0;
        wS[tid]   = v ? gw[e * N_TOK + idx]   : 0.f;
    }
    __syncthreads();

    const int wave = tid >> 5, lane = tid & 31;
    const int lane_lo = lane & 15, lane_hi = lane >> 4;
    const int mw = wave >> 2, fw = wave & 3;     // 2 (M) x 4 (F-pairs) wave grid

    v8f acc0 = {};
    v8f acc1 = {};
    const float* wbase = ew + (size_t)e * DIM * DIM;

    // per-thread staging coordinates (hoisted):
    // X: 32 rows x 8 float4 = 256 b128 -> 1 per thread
    const int xrow = tid >> 3, xc = (tid & 7) << 2;
    const float* xptr = x + (size_t)tokS[xrow] * DIM + xc;
    const unsigned xl[2] = { (unsigned)(uintptr_t)&Xs[0][xrow * XST + xc],
                             (unsigned)(uintptr_t)&Xs[1][xrow * XST + xc] };
    // W: 128 rows x 8 float4 = 1024 b128 -> 4 per thread
    const float* wptr[4];
    unsigned wl[2][4];
    #pragma unroll
    for (int i = 0; i < 4; ++i) {
        const int v = tid + 256 * i;
        const int row = v >> 3, c = (v & 7) << 2;
        wptr[i]  = wbase + (size_t)(f0 + row) * DIM + c;
        wl[0][i] = (unsigned)(uintptr_t)&Ws[0][row * XST + c];
        wl[1][i] = (unsigned)(uintptr_t)&Ws[1][row * XST + c];
    }

    auto fill = [&](int buf, int kc) {
        async_g2l_b128(xl[buf], xptr + kc);
        #pragma unroll
        for (int i = 0; i < 4; ++i)
            async_g2l_b128(wl[buf][i], wptr[i] + kc);
    };

    fill(0, 0);                                    // prologue

    const int NCHUNK = DIM / KC;                   // 32
    for (int chunk = 0; chunk < NCHUNK; ++chunk) {
        const int buf = chunk & 1;
        wait_async0();          // my in-flight fill of `buf` complete
        __syncthreads();        // all fills of `buf` visible; all reads of `buf^1` done
        if (chunk + 1 < NCHUNK)
            fill(buf ^ 1, (chunk + 1) * KC);       // overlaps compute below

        // A: lane holds X[m = lane_lo][k = 4s + 2*lane_hi + {0,1}]
        // B: lane holds W[n = lane_lo][k = 4s + 2*lane_hi + {0,1}]  (B = W^T, K-major per lane)
        const float* xa  = &Xs[buf][(mw * 16 + lane_lo) * XST + 2 * lane_hi];
        const float* wb0 = &Ws[buf][(fw * 32 + lane_lo) * XST + 2 * lane_hi];
        const float* wb1 = wb0 + 16 * XST;
        #pragma unroll
        for (int s = 0; s < KC / 4; ++s) {
            v2f a  = *(const v2f*)(xa  + 4 * s);
            v2f b0 = *(const v2f*)(wb0 + 4 * s);
            v2f b1 = *(const v2f*)(wb1 + 4 * s);
            acc0 = __builtin_amdgcn_wmma_f32_16x16x4_f32(
                false, a, false, b0, (short)0, acc0, false, false);
            acc1 = __builtin_amdgcn_wmma_f32_16x16x4_f32(
                false, a, false, b1, (short)0, acc1, false, false);
        }
        // no trailing barrier: next iteration's barrier protects `buf^1`'s refill,
        // and `buf` is only refilled after the barrier one iteration later.
    }

    // epilogue: D layout -> VGPR r, lanes 0-15: M=r, lanes 16-31: M=8+r; N=lane_lo
    const int f = f0 + fw * 32 + lane_lo;
    const float bias0 = eb[e * DIM + f];
    const float bias1 = eb[e * DIM + f + 16];
    #pragma unroll
    for (int r = 0; r < 8; ++r) {
        const int ml = mw * 16 + r + 8 * lane_hi;
        if (m0 + ml < count) {
            const float g = wS[ml];
            float* orow = out + (size_t)tokS[ml] * DIM;
            atomicAdd(orow + f,      g * (acc0[r] + bias0));
            atomicAdd(orow + f + 16, g * (acc1[r] + bias1));
        }
    }
}

// ---------------- Load-balancing loss ----------------
__global__ void moe_loss(const int* __restrict__ cnt, const float* __restrict__ wsum,
                         float* __restrict__ out)
{
    if (threadIdx.x == 0 && blockIdx.x == 0) {
        const float invN = 1.f / (float)N_TOK;
        float loss = 0.f;
        for (int e = 0; e < NEXP; ++e)
            loss += (wsum[e] * invN) * ((float)cnt[e] * invN);
        out[(size_t)N_TOK * DIM] = loss;
    }
}

extern "C" void kernel_launch(void* const* d_in, const int* in_sizes, int n_in,
                              void* d_out, int out_size, void* d_ws, size_t ws_size,
                              hipStream_t stream) {
    const float* x   = (const float*)d_in[0];
    const float* ctx = (const float*)d_in[1];
    const float* rw  = (const float*)d_in[2];
    const float* rb  = (const float*)d_in[3];
    const float* ew  = (const float*)d_in[4];
    const float* eb  = (const float*)d_in[5];
    const float* cw  = (const float*)d_in[6];
    float* out = (float*)d_out;

    char*  ws   = (char*)d_ws;
    int*   cnt  = (int*)ws;                                      // 8 ints
    float* wsum = (float*)(ws + 32);                             // 8 floats
    int*   list = (int*)(ws + 64);                               // 8*N ints
    float* gw   = (float*)(ws + 64 + (size_t)NEXP * N_TOK * 4);  // 8*N floats

    hipMemsetAsync(ws, 0, 64, stream);
    hipMemsetAsync(out, 0, (size_t)N_TOK * DIM * sizeof(float), stream);

    moe_router<<<N_TOK / 64, 256, 0, stream>>>(x, ctx, rw, rb, cw, cnt, wsum, list, gw);

    dim3 g2(DIM / FT, N_TOK / MT, NEXP);   // (8, 512, 8); most blocks exit on count check
    moe_expert_gemm<<<g2, 256, 0, stream>>>(x, ew, eb, cnt, list, gw, out);

    moe_loss<<<1, 32, 0, stream>>>(cnt, wsum, out);
}